// LearnableFF_32736240730169
// MI455X (gfx1250) — compile-verified
//
#include <hip/hip_runtime.h>
#include <cstdint>

typedef __attribute__((ext_vector_type(2))) float v2f;
typedef __attribute__((ext_vector_type(4))) float v4f;
typedef __attribute__((ext_vector_type(8))) float v8f;

#define L_ATOMS 1024
#define R_ATOMS 8192
#define FEAT    64
#define STEPS   8

// workspace layout (float offsets)
#define WS_ATN   0                               // 1024*8192 = 8388608 floats (32MB)
#define WS_LIGC  (L_ATOMS * R_ATOMS)             // centered lig coords, 3072
#define WS_RECC  (WS_LIGC + L_ATOMS * 3)         // centered rec coords, 24576
#define WS_PARAM (WS_RECC + R_ATOMS * 3)         // params: [0..8]=pr [9..11]=tr [12..20]=Q [21..29]=R
#define WS_GPART (WS_PARAM + 32)                 // 128 blocks * 12 gradient partials

// ---------------------------------------------------------------------------
// Preprocess: center coordinates, copy pr, scale trans by TRANS_DIST=4.
// ---------------------------------------------------------------------------
__global__ __launch_bounds__(256) void ff_preproc(
    const float* __restrict__ lig_in, const float* __restrict__ rec_in,
    const float* __restrict__ pre_rot, const float* __restrict__ trans,
    float* __restrict__ ligc, float* __restrict__ recc, float* __restrict__ prm) {
  __shared__ float red[768];
  const int tid = threadIdx.x;

  float sx = 0.f, sy = 0.f, sz = 0.f;
  for (int i = tid; i < L_ATOMS; i += 256) {
    sx += lig_in[3 * i]; sy += lig_in[3 * i + 1]; sz += lig_in[3 * i + 2];
  }
  red[tid] = sx; red[256 + tid] = sy; red[512 + tid] = sz;
  __syncthreads();
  for (int s = 128; s > 0; s >>= 1) {
    if (tid < s) {
      red[tid] += red[tid + s];
      red[256 + tid] += red[256 + tid + s];
      red[512 + tid] += red[512 + tid + s];
    }
    __syncthreads();
  }
  const float mlx = red[0] * (1.f / L_ATOMS);
  const float mly = red[256] * (1.f / L_ATOMS);
  const float mlz = red[512] * (1.f / L_ATOMS);
  __syncthreads();

  sx = sy = sz = 0.f;
  for (int i = tid; i < R_ATOMS; i += 256) {
    sx += rec_in[3 * i]; sy += rec_in[3 * i + 1]; sz += rec_in[3 * i + 2];
  }
  red[tid] = sx; red[256 + tid] = sy; red[512 + tid] = sz;
  __syncthreads();
  for (int s = 128; s > 0; s >>= 1) {
    if (tid < s) {
      red[tid] += red[tid + s];
      red[256 + tid] += red[256 + tid + s];
      red[512 + tid] += red[512 + tid + s];
    }
    __syncthreads();
  }
  const float mrx = red[0] * (1.f / R_ATOMS);
  const float mry = red[256] * (1.f / R_ATOMS);
  const float mrz = red[512] * (1.f / R_ATOMS);

  for (int i = tid; i < L_ATOMS; i += 256) {
    ligc[3 * i]     = lig_in[3 * i]     - mlx;
    ligc[3 * i + 1] = lig_in[3 * i + 1] - mly;
    ligc[3 * i + 2] = lig_in[3 * i + 2] - mlz;
  }
  for (int i = tid; i < R_ATOMS; i += 256) {
    recc[3 * i]     = rec_in[3 * i]     - mrx;
    recc[3 * i + 1] = rec_in[3 * i + 1] - mry;
    recc[3 * i + 2] = rec_in[3 * i + 2] - mrz;
  }
  if (tid == 0) {
    for (int i = 0; i < 9; ++i) prm[i] = pre_rot[i];
    for (int i = 0; i < 3; ++i) prm[9 + i] = trans[i] * 4.0f;  // TRANS_DIST
  }
}

// ---------------------------------------------------------------------------
// atn = lig_feat @ rec_feat^T via V_WMMA_F32_16X16X4_F32 (fp32 tensor path).
// One wave per 16x16 C tile, 16 WMMAs over K=64. Block = 8 waves.
// ---------------------------------------------------------------------------
__global__ __launch_bounds__(256) void ff_gemm_wmma(
    const float* __restrict__ A,   // lig_feat [1024,64]
    const float* __restrict__ B,   // rec_feat [8192,64]
    float* __restrict__ C) {       // atn [1024,8192]
  const int wave = threadIdx.x >> 5;
  const int lane = threadIdx.x & 31;
  const int half = lane >> 4;      // selects K pair (0/1 vs 2/3)
  const int m    = lane & 15;      // M for A, N for B
  const int l0 = blockIdx.y * 16;
  const int r0 = (blockIdx.x * 8 + wave) * 16;

  const float* arow = A + (size_t)(l0 + m) * FEAT + 2 * half;
  const float* brow = B + (size_t)(r0 + m) * FEAT + 2 * half;

  v8f c = {0.f, 0.f, 0.f, 0.f, 0.f, 0.f, 0.f, 0.f};
#pragma unroll
  for (int kc = 0; kc < 16; ++kc) {
    v2f av = *(const v2f*)(arow + 4 * kc);  // A[m, 4kc + 2*half + {0,1}]
    v2f bv = *(const v2f*)(brow + 4 * kc);  // B^T: rec_feat[r0+m, 4kc + 2*half + {0,1}]
    c = __builtin_amdgcn_wmma_f32_16x16x4_f32(false, av, false, bv,
                                              (short)0, c, false, false);
  }
  // C layout: VGPR i -> row M = i + 8*half, col N = m
  float* crow = C + (size_t)(l0 + 8 * half) * R_ATOMS + (r0 + m);
#pragma unroll
  for (int i = 0; i < 8; ++i) crow[(size_t)i * R_ATOMS] = c[i];
}

// ---------------------------------------------------------------------------
// Householder QR of 3x3 pr (LAPACK sign convention: diag(R) = -sign(pivot)*norm)
// Writes Q -> prm[12..20], R -> prm[21..29]. Single thread.
// ---------------------------------------------------------------------------
__global__ void ff_qr(float* __restrict__ prm) {
  if (threadIdx.x != 0 || blockIdx.x != 0) return;
  float A[3][3], Q[3][3];
  for (int i = 0; i < 3; ++i)
    for (int j = 0; j < 3; ++j) {
      A[i][j] = prm[3 * i + j];
      Q[i][j] = (i == j) ? 1.f : 0.f;
    }
  for (int k = 0; k < 2; ++k) {
    float nx = 0.f;
    for (int i = k; i < 3; ++i) nx += A[i][k] * A[i][k];
    nx = sqrtf(nx);
    const float alpha = A[k][k];
    const float beta = (alpha >= 0.f) ? -nx : nx;
    float v[3] = {0.f, 0.f, 0.f};
    v[k] = alpha - beta;
    for (int i = k + 1; i < 3; ++i) v[i] = A[i][k];
    float vtv = 0.f;
    for (int i = k; i < 3; ++i) vtv += v[i] * v[i];
    if (vtv > 0.f) {
      const float s = 2.f / vtv;
      for (int j = 0; j < 3; ++j) {           // A = H A
        float d = 0.f;
        for (int i = k; i < 3; ++i) d += v[i] * A[i][j];
        d *= s;
        for (int i = k; i < 3; ++i) A[i][j] -= v[i] * d;
      }
      for (int i = 0; i < 3; ++i) {           // Q = Q H
        float d = 0.f;
        for (int j = k; j < 3; ++j) d += Q[i][j] * v[j];
        d *= s;
        for (int j = k; j < 3; ++j) Q[i][j] -= v[j] * d;
      }
    }
  }
  for (int i = 0; i < 3; ++i)
    for (int j = 0; j < 3; ++j) {
      prm[12 + 3 * i + j] = Q[i][j];
      prm[21 + 3 * i + j] = (j >= i) ? A[i][j] : 0.f;
    }
}

// ---------------------------------------------------------------------------
// Main step: one wave per ligand row l. Computes new_lig row (output), then
// reduces w = atn/d2^2 statistics over r=0..8191. Receptor coords staged in
// LDS via GLOBAL_LOAD_ASYNC_TO_LDS_B128 (ASYNCcnt); atn streamed as b128.
// Deterministic: wave shuffle reduce -> LDS -> per-block partial (no atomics).
// ---------------------------------------------------------------------------
__global__ __launch_bounds__(256) void ff_step(
    const float* __restrict__ atn, const float* __restrict__ ligc,
    const float* __restrict__ recc, const float* __restrict__ prm,
    float* __restrict__ out_step, float* __restrict__ gpart) {
  __shared__ __align__(16) float srec[2048 * 3];   // 24 KB chunk of rec coords
  __shared__ float part[8][12];
  const int tid = threadIdx.x;
  const int wave = tid >> 5;
  const int lane = tid & 31;
  const int l = blockIdx.x * 8 + wave;

  const float* rot = prm + 12;
  const float* tr  = prm + 9;
  const float lx = ligc[3 * l], ly = ligc[3 * l + 1], lz = ligc[3 * l + 2];
  const float nl0 = rot[0] * lx + rot[1] * ly + rot[2] * lz + tr[0];
  const float nl1 = rot[3] * lx + rot[4] * ly + rot[5] * lz + tr[1];
  const float nl2 = rot[6] * lx + rot[7] * ly + rot[8] * lz + tr[2];

  // LDS byte offset of srec (generic ptr: high 32 bits = shared aperture)
  const uint32_t lds_base = (uint32_t)(uintptr_t)(void*)&srec[0];

  float aw = 0.f, ax = 0.f, ay = 0.f, az = 0.f;
  const float* arow = atn + (size_t)l * R_ATOMS;

  for (int chunk = 0; chunk < R_ATOMS; chunk += 2048) {
    __syncthreads();
    // async copy 2048*3 floats = 1536 x b128 straight into LDS
    {
      const uint64_t gsrc = (uint64_t)(uintptr_t)(recc + (size_t)chunk * 3);
      for (int i = tid; i < 1536; i += 256) {
        const uint32_t ldsa = lds_base + (uint32_t)(i * 16);
        const uint64_t ga = gsrc + (uint64_t)((size_t)i * 16);
        asm volatile("global_load_async_to_lds_b128 %0, %1, off"
                     :: "v"(ldsa), "v"(ga) : "memory");
      }
      asm volatile("s_wait_asynccnt 0x0" ::: "memory");
    }
    __syncthreads();
    if (chunk + 2048 < R_ATOMS)  // prefetch next atn chunk (global_prefetch_b8)
      __builtin_prefetch(arow + chunk + 2048 + lane * 16, 0, 3);

    for (int rb = lane * 4; rb < 2048; rb += 128) {
      const v4f a4 = *(const v4f*)(arow + chunk + rb);
      const v4f c0 = *(const v4f*)(srec + 3 * rb);       // 48B/lane, 16B aligned
      const v4f c1 = *(const v4f*)(srec + 3 * rb + 4);
      const v4f c2 = *(const v4f*)(srec + 3 * rb + 8);
      const float rx[4] = {c0[0], c0[3], c1[2], c2[1]};
      const float ry[4] = {c0[1], c1[0], c1[3], c2[2]};
      const float rz[4] = {c0[2], c1[1], c2[0], c2[3]};
#pragma unroll
      for (int j = 0; j < 4; ++j) {
        const float dx = nl0 - rx[j], dy = nl1 - ry[j], dz = nl2 - rz[j];
        const float d2 = dx * dx + dy * dy + dz * dz;
        const float inv = 1.0f / d2;
        const float w = a4[j] * inv * inv;
        aw += w; ax += w * rx[j]; ay += w * ry[j]; az += w * rz[j];
      }
    }
  }
  // wave32 butterfly reduction
  for (int off = 16; off > 0; off >>= 1) {
    aw += __shfl_xor(aw, off, 32);
    ax += __shfl_xor(ax, off, 32);
    ay += __shfl_xor(ay, off, 32);
    az += __shfl_xor(az, off, 32);
  }
  if (lane == 0) {
    out_step[3 * l] = nl0; out_step[3 * l + 1] = nl1; out_step[3 * l + 2] = nl2;
    const float coef = -2.0f / ((float)L_ATOMS * (float)R_ATOMS);
    const float g0 = coef * (aw * nl0 - ax);
    const float g1 = coef * (aw * nl1 - ay);
    const float g2 = coef * (aw * nl2 - az);
    part[wave][0] = g0;  part[wave][1] = g1;  part[wave][2] = g2;       // g_tr
    part[wave][3] = g0 * lx; part[wave][4] = g0 * ly; part[wave][5] = g0 * lz;  // g_rot row 0
    part[wave][6] = g1 * lx; part[wave][7] = g1 * ly; part[wave][8] = g1 * lz;  // row 1
    part[wave][9] = g2 * lx; part[wave][10] = g2 * ly; part[wave][11] = g2 * lz; // row 2
  }
  __syncthreads();
  if (tid < 12) {
    float s = 0.f;
    for (int w = 0; w < 8; ++w) s += part[w][tid];
    gpart[blockIdx.x * 12 + tid] = s;
  }
}

// ---------------------------------------------------------------------------
// Final gradient reduce + QR VJP (R_bar = 0) + parameter update. Single thread.
//   B = Q^T Gq ;  g_pr = Q * stricttril(B - B^T) * R^{-T}
//   pr -= g_pr (ROT_LR=1) ; tr -= 0.1 * g_tr (TRANS_LR)
// ---------------------------------------------------------------------------
__global__ void ff_update(float* __restrict__ prm, const float* __restrict__ gpart) {
  if (threadIdx.x != 0 || blockIdx.x != 0) return;
  float g[12];
  for (int k = 0; k < 12; ++k) {
    float s = 0.f;
    for (int b = 0; b < 128; ++b) s += gpart[b * 12 + k];
    g[k] = s;
  }
  float Gq[3][3] = {{g[3], g[4], g[5]}, {g[6], g[7], g[8]}, {g[9], g[10], g[11]}};
  float Q[3][3], R[3][3];
  for (int i = 0; i < 3; ++i)
    for (int j = 0; j < 3; ++j) {
      Q[i][j] = prm[12 + 3 * i + j];
      R[i][j] = prm[21 + 3 * i + j];
    }
  float Bm[3][3];
  for (int i = 0; i < 3; ++i)
    for (int j = 0; j < 3; ++j) {
      float s = 0.f;
      for (int k = 0; k < 3; ++k) s += Q[k][i] * Gq[k][j];
      Bm[i][j] = s;
    }
  float S[3][3] = {};
  for (int i = 0; i < 3; ++i)
    for (int j = 0; j < i; ++j) S[i][j] = Bm[i][j] - Bm[j][i];
  float T[3][3];
  for (int i = 0; i < 3; ++i)
    for (int j = 0; j < 3; ++j) {
      float s = 0.f;
      for (int k = 0; k < 3; ++k) s += Q[i][k] * S[k][j];
      T[i][j] = s;
    }
  // solve X * R^T = T  (R upper-triangular): back-substitute per row
  float X[3][3];
  for (int i = 0; i < 3; ++i) {
    X[i][2] = T[i][2] / R[2][2];
    X[i][1] = (T[i][1] - X[i][2] * R[1][2]) / R[1][1];
    X[i][0] = (T[i][0] - X[i][1] * R[0][1] - X[i][2] * R[0][2]) / R[0][0];
  }
  for (int i = 0; i < 3; ++i)
    for (int j = 0; j < 3; ++j) prm[3 * i + j] -= X[i][j];   // ROT_LR = 1.0
  prm[9]  -= 0.1f * g[0];
  prm[10] -= 0.1f * g[1];
  prm[11] -= 0.1f * g[2];
}

// ---------------------------------------------------------------------------
extern "C" void kernel_launch(void* const* d_in, const int* in_sizes, int n_in,
                              void* d_out, int out_size, void* d_ws, size_t ws_size,
                              hipStream_t stream) {
  (void)in_sizes; (void)n_in; (void)out_size; (void)ws_size;
  const float* lig_feat  = (const float*)d_in[0];   // [1024,64]
  const float* rec_feat  = (const float*)d_in[1];   // [8192,64]
  const float* lig_coord = (const float*)d_in[2];   // [1024,3]
  const float* rec_coord = (const float*)d_in[3];   // [8192,3]
  const float* pre_rot   = (const float*)d_in[4];   // [3,3]
  const float* trans     = (const float*)d_in[5];   // [3]
  float* out = (float*)d_out;                       // [8,1024,3]
  float* ws  = (float*)d_ws;

  float* atn   = ws + WS_ATN;
  float* ligc  = ws + WS_LIGC;
  float* recc  = ws + WS_RECC;
  float* prm   = ws + WS_PARAM;
  float* gpart = ws + WS_GPART;

  ff_preproc<<<1, 256, 0, stream>>>(lig_coord, rec_coord, pre_rot, trans,
                                    ligc, recc, prm);
  ff_gemm_wmma<<<dim3(R_ATOMS / 128, L_ATOMS / 16), 256, 0, stream>>>(
      lig_feat, rec_feat, atn);

  for (int s = 0; s < STEPS; ++s) {
    ff_qr<<<1, 1, 0, stream>>>(prm);
    ff_step<<<L_ATOMS / 8, 256, 0, stream>>>(atn, ligc, recc, prm,
                                             out + (size_t)s * L_ATOMS * 3, gpart);
    ff_update<<<1, 1, 0, stream>>>(prm, gpart);
  }
}